// Net_24275155157688
// MI455X (gfx1250) — compile-verified
//
#include <hip/hip_runtime.h>

typedef __attribute__((ext_vector_type(2))) float v2f;
typedef __attribute__((ext_vector_type(8))) float v8f;

#define GN 64          // graph size N
#define LSTR 68        // padded LDS row stride (floats): avoids bank conflicts, keeps 16B align
#define NTHREADS 512   // 16 wave32 waves -> 4x4 grid of 16x16 output tiles

__global__ __launch_bounds__(NTHREADS) void trace_series_fc(
    const float* __restrict__ x,   // [B, 64, 64]
    const float* __restrict__ W,   // [2, 32]
    float* __restrict__ out) {     // [B, 2]
  __shared__ float sX[GN * LSTR];  // x
  __shared__ float sA[GN * LSTR];  // x^2
  __shared__ float sM[GN * LSTR];  // x^3
  __shared__ float rowA[GN];
  __shared__ float rowM[GN];
  __shared__ float red[8];         // tr2,tr3,tr4,tr5,s2,s3,s4,s5

  const int tid = threadIdx.x;
  const int b = blockIdx.x;
  const float* xb = x + (size_t)b * GN * GN;

  // ---- cooperative load of x tile into LDS (float4, 2 iters/thread) ----
  for (int t = tid; t < GN * GN / 4; t += NTHREADS) {
    const float4 v = reinterpret_cast<const float4*>(xb)[t];
    const int idx = t * 4;
    const int r = idx >> 6;
    const int c = idx & 63;
    *reinterpret_cast<float4*>(&sX[r * LSTR + c]) = v;
  }
  __syncthreads();

  const int lane = tid & 31;
  const int wave = tid >> 5;
  const int half = lane >> 4;      // 0: K 0..1, 1: K 2..3 within each K=4 step
  const int ml   = lane & 15;      // row (A-frag) / col (B,C,D-frags)
  const int tm   = (wave >> 2) * 16;  // output tile row base
  const int tn   = (wave & 3) * 16;   // output tile col base

  // ---- A = x * x (x^2), each wave one 16x16 tile, K=64 in 16 WMMA steps ----
  {
    v8f acc = {};
    for (int kk = 0; kk < GN; kk += 4) {
      const int ka = kk + half * 2;
      v2f af, bf;
      af.x = sX[(tm + ml) * LSTR + ka];
      af.y = sX[(tm + ml) * LSTR + ka + 1];
      bf.x = sX[ka * LSTR + tn + ml];
      bf.y = sX[(ka + 1) * LSTR + tn + ml];
      acc = __builtin_amdgcn_wmma_f32_16x16x4_f32(
          false, af, false, bf, (short)0, acc, false, false);
    }
#pragma unroll
    for (int r = 0; r < 8; ++r)
      sA[(tm + r + 8 * half) * LSTR + tn + ml] = acc[r];
  }
  __syncthreads();

  // ---- M = x * A (x^3) ----
  {
    v8f acc = {};
    for (int kk = 0; kk < GN; kk += 4) {
      const int ka = kk + half * 2;
      v2f af, bf;
      af.x = sX[(tm + ml) * LSTR + ka];
      af.y = sX[(tm + ml) * LSTR + ka + 1];
      bf.x = sA[ka * LSTR + tn + ml];
      bf.y = sA[(ka + 1) * LSTR + tn + ml];
      acc = __builtin_amdgcn_wmma_f32_16x16x4_f32(
          false, af, false, bf, (short)0, acc, false, false);
    }
#pragma unroll
    for (int r = 0; r < 8; ++r)
      sM[(tm + r + 8 * half) * LSTR + tn + ml] = acc[r];
  }
  __syncthreads();

  // ---- row sums of A and M (for sum(x^4), sum(x^5)) ----
  if (tid < GN) {
    float ra = 0.f, rm = 0.f;
    for (int c = 0; c < GN; ++c) {
      ra += sA[tid * LSTR + c];
      rm += sM[tid * LSTR + c];
    }
    rowA[tid] = ra;
    rowM[tid] = rm;
  }
  if (tid < 8) red[tid] = 0.f;
  __syncthreads();

  // ---- O(N^2) scalar reductions: 8 elements per thread ----
  float tr2 = 0.f, tr3 = 0.f, tr4 = 0.f, tr5 = 0.f;
  float s2 = 0.f, s3 = 0.f, s4 = 0.f, s5 = 0.f;
  for (int t = tid; t < GN * GN; t += NTHREADS) {
    const int i = t >> 6;
    const int j = t & 63;
    const float a  = sA[i * LSTR + j];
    const float m  = sM[i * LSTR + j];
    const float at = sA[j * LSTR + i];
    const float mt = sM[j * LSTR + i];
    s2  += a;                 // sum(x^2)
    s3  += m;                 // sum(x^3)
    tr4 += a * at;            // tr(x^4) = <A, A^T>
    tr5 += a * mt;            // tr(x^5) = <A, M^T>
    s4  += a * rowA[j];       // sum(x^4) = sum_ij A_ij * rowA_j
    s5  += a * rowM[j];       // sum(x^5) = sum_ij A_ij * rowM_j
    if (i == j) { tr2 += a; tr3 += m; }
  }
  // wave32 tree reduction, then cross-wave via LDS float atomics (ds_add_f32)
#pragma unroll
  for (int off = 16; off > 0; off >>= 1) {
    tr2 += __shfl_xor(tr2, off);
    tr3 += __shfl_xor(tr3, off);
    tr4 += __shfl_xor(tr4, off);
    tr5 += __shfl_xor(tr5, off);
    s2  += __shfl_xor(s2, off);
    s3  += __shfl_xor(s3, off);
    s4  += __shfl_xor(s4, off);
    s5  += __shfl_xor(s5, off);
  }
  if (lane == 0) {
    atomicAdd(&red[0], tr2);
    atomicAdd(&red[1], tr3);
    atomicAdd(&red[2], tr4);
    atomicAdd(&red[3], tr5);
    atomicAdd(&red[4], s2);
    atomicAdd(&red[5], s3);
    atomicAdd(&red[6], s4);
    atomicAdd(&red[7], s5);
  }
  __syncthreads();

  // ---- features + 32->2 linear layer (tiny; single thread) ----
  if (tid == 0) {
    float tr[4] = {red[0], red[1], red[2], red[3]};
    float ss[4] = {red[4], red[5], red[6], red[7]};
    float feat[32];
#pragma unroll
    for (int i = 0; i < 4; ++i) {
      float tp = 1.f, sp = 1.f;
#pragma unroll
      for (int j = 0; j < 4; ++j) {
        tp *= tr[i];  // tr_i^(j+1)
        sp *= ss[i];  // s_i^(j+1)
        // numel = 4096 = 2^12 ; numel^e = 2^(12e)
        const float den_tr   = __builtin_ldexpf(1.0f, 12 * (i + j + 1));
        const float den_full = __builtin_ldexpf(1.0f, 12 * (i + j + 2));
        feat[i * 4 + j]      = tp / den_tr;
        feat[16 + i * 4 + j] = sp / den_full;
      }
    }
    float o0 = 0.f, o1 = 0.f;
#pragma unroll
    for (int f = 0; f < 32; ++f) {
      o0 += feat[f] * W[f];
      o1 += feat[f] * W[32 + f];
    }
    out[b * 2 + 0] = o0;
    out[b * 2 + 1] = o1;
  }
}

extern "C" void kernel_launch(void* const* d_in, const int* in_sizes, int n_in,
                              void* d_out, int out_size, void* d_ws, size_t ws_size,
                              hipStream_t stream) {
  const float* x = (const float*)d_in[0];
  const float* W = (const float*)d_in[1];
  float* out = (float*)d_out;
  const int B = in_sizes[0] / (GN * GN);
  trace_series_fc<<<dim3(B), dim3(NTHREADS), 0, stream>>>(x, W, out);
}